// GeneralScore_44367012168321
// MI455X (gfx1250) — compile-verified
//
#include <hip/hip_runtime.h>

typedef __attribute__((ext_vector_type(2))) float v2f;
typedef __attribute__((ext_vector_type(8))) float v8f;

#define Hdim 1024
#define Tdim 4096
#define Bdim 8

// ---------------------------------------------------------------------------
// Phase 1: v^T (H x 8) = W (H x H) @ context^T (H x 8), N padded to 16.
// One wave (32 lanes) per 16-row tile of W. K loop: 256 x V_WMMA_F32_16X16X4_F32.
//
// A (16x4 f32) lane layout (ISA 7.12.2): lane L holds M = L%16,
//   VGPR0/1 = K = (L/16)*2 + {0,1}.
// B (4x16 f32): VGPR0/1 = row K = (L/16)*2 + {0,1}, N = L%16.
// C/D (16x16 f32): VGPR r: lanes 0-15 -> M=r, lanes 16-31 -> M=r+8, N = L%16.
// Columns N>=8 of D are never stored, so feeding duplicated context rows for
// those columns (clamped index, keeps EXEC uniform) is harmless.
// ---------------------------------------------------------------------------
__global__ __launch_bounds__(32) void bilinear_v_wmma(const float* __restrict__ W,
                                                      const float* __restrict__ ctx,
                                                      float* __restrict__ v) {
  const int lane = threadIdx.x;            // 0..31
  const int m0   = blockIdx.x * 16;        // row tile of W / of v^T
  const int mA   = lane & 15;
  const int kHi  = (lane >> 4) * 2;        // 0 or 2
  const int n    = lane & 15;              // output column (batch index)
  const int nc   = (n < Bdim) ? n : (Bdim - 1);  // clamp: keep EXEC all-ones

  const float* wrow = W   + (size_t)(m0 + mA) * Hdim + kHi;  // A operand stream
  const float* crow = ctx + (size_t)nc * Hdim + kHi;         // B operand stream

  v8f acc = {};
  for (int k0 = 0; k0 < Hdim; k0 += 4) {
    v2f a, b;
    a.x = wrow[k0];
    a.y = wrow[k0 + 1];
    b.x = crow[k0];
    b.y = crow[k0 + 1];
    // D = A x B + C   (f32, exact — matches reference precision)
    acc = __builtin_amdgcn_wmma_f32_16x16x4_f32(
        /*neg_a=*/false, a, /*neg_b=*/false, b,
        /*c_mod=*/(short)0, acc, /*reuse_a=*/false, /*reuse_b=*/false);
  }

  // D[m][n] = sum_k W[m0+m,k] * ctx[n,k] = v[n, m0+m]; store real columns only.
  if (n < Bdim) {
    const int mBase = m0 + ((lane >> 4) * 8);
#pragma unroll
    for (int r = 0; r < 8; ++r) {
      v[(size_t)n * Hdim + mBase + r] = acc[r];
    }
  }
}

// ---------------------------------------------------------------------------
// Phase 2: scores[b,t] = dot(states[b,t,:], v[b,:]) + bias.
// Pure bandwidth phase: 134 MB of states read once. One wave per row; lanes
// split H so every load is a coalesced float4 (global_load_b128, 512B/wave).
// v[b] lives in 32 VGPRs per lane, loaded once per wave, reused for 8 rows.
// Wave32 reduction via __shfl_xor (5 steps).
// ---------------------------------------------------------------------------
__global__ __launch_bounds__(256) void score_gemv(const float* __restrict__ states,
                                                  const float* __restrict__ v,
                                                  const float* __restrict__ bias,
                                                  float* __restrict__ out,
                                                  int wavesPerBatch) {
  const int lane        = threadIdx.x & 31;
  const int waveInBlock = threadIdx.x >> 5;
  const int waveId      = blockIdx.x * (blockDim.x >> 5) + waveInBlock;
  const int b           = waveId / wavesPerBatch;   // 0..7
  const int tw          = waveId % wavesPerBatch;

  // Lane 'l' owns columns i*128 + l*4 .. +3 for i in 0..7 (covers H=1024).
  const float4* vb = reinterpret_cast<const float4*>(v + (size_t)b * Hdim);
  float4 vr[8];
#pragma unroll
  for (int i = 0; i < 8; ++i) vr[i] = vb[i * 32 + lane];

  const float bias0 = bias[0];

  for (int t = tw; t < Tdim; t += wavesPerBatch) {
    const float4* srow =
        reinterpret_cast<const float4*>(states + ((size_t)b * Tdim + t) * Hdim);
    float acc = 0.f;
#pragma unroll
    for (int i = 0; i < 8; ++i) {
      float4 s = srow[i * 32 + lane];
      acc += s.x * vr[i].x + s.y * vr[i].y + s.z * vr[i].z + s.w * vr[i].w;
    }
#pragma unroll
    for (int off = 16; off > 0; off >>= 1) acc += __shfl_xor(acc, off, 32);
    if (lane == 0) out[(size_t)b * Tdim + t] = acc + bias0;
  }
}

// ---------------------------------------------------------------------------
// Inputs (setup_inputs order): states (B,T,H) f32, context (B,H) f32,
// W (1,H,H) f32, b (1,) f32. Output: (B,T,1) f32.
// Workspace: v (B*H floats = 32 KB) in d_ws.
// ---------------------------------------------------------------------------
extern "C" void kernel_launch(void* const* d_in, const int* in_sizes, int n_in,
                              void* d_out, int out_size, void* d_ws, size_t ws_size,
                              hipStream_t stream) {
  const float* states = (const float*)d_in[0];
  const float* ctx    = (const float*)d_in[1];
  const float* W      = (const float*)d_in[2];
  const float* bias   = (const float*)d_in[3];
  float*       out    = (float*)d_out;
  float*       v      = (float*)d_ws;   // B*H f32 scratch

  // Phase 1: 64 tiles of 16 rows, one wave each.
  bilinear_v_wmma<<<Hdim / 16, 32, 0, stream>>>(W, ctx, v);

  // Phase 2: 4096 waves total; 512 waves per batch -> 8 rows per wave.
  const int wavesPerBatch = 512;               // (B*T) / totalWaves = 8 rows/wave
  score_gemv<<<512, 256, 0, stream>>>(states, v, bias, out, wavesPerBatch);
}